// GraphAttentionLayer_2611340116281
// MI455X (gfx1250) — compile-verified
//
#include <hip/hip_runtime.h>
#include <stdint.h>

typedef __attribute__((ext_vector_type(2))) float v2f;
typedef __attribute__((ext_vector_type(8))) float v8f;

#define HEADS 4
#define NEG_SLOPE 0.2f
#define RSQRT_D 0.17677669529663687f  // 1/sqrt(32)

// Ordered-uint mapping for float atomic max (monotone for all finite floats)
__device__ __forceinline__ unsigned fmap(float f) {
  unsigned b = __float_as_uint(f);
  return (b & 0x80000000u) ? ~b : (b | 0x80000000u);
}
__device__ __forceinline__ float funmap(unsigned u) {
  return __uint_as_float((u & 0x80000000u) ? (u ^ 0x80000000u) : ~u);
}

// ---------------------------------------------------------------------------
// Zero the accumulator (aliased with d_out) and reset softmax reduction state.
// ---------------------------------------------------------------------------
__global__ void init_kernel(float* __restrict__ acc, long long total,
                            unsigned* __restrict__ gmax, float* __restrict__ sums) {
  long long i = (long long)blockIdx.x * blockDim.x + threadIdx.x;
  long long stride = (long long)gridDim.x * blockDim.x;
  for (; i < total; i += stride) acc[i] = 0.0f;
  if (blockIdx.x == 0 && threadIdx.x < HEADS) {
    gmax[threadIdx.x] = 0u;       // fmap(-inf) < anything finite
    sums[threadIdx.x] = 0.0f;
  }
}

// ---------------------------------------------------------------------------
// Y[N,128] = X[N,128] @ W[128,128] + bias, fp32 WMMA 16x16x4.
// Block = 256 threads = 8 waves; block handles 16 rows, wave w handles cols
// 16w..16w+15. A-tile staged via LDS so Y may alias X (in-place final GEMM):
// each row block is read only by its own workgroup, and only before stores.
// ---------------------------------------------------------------------------
__global__ __launch_bounds__(256) void gemm128_wmma(
    const float* __restrict__ X, const float* __restrict__ W,
    const float* __restrict__ bias, float* __restrict__ Y, int N) {
  __shared__ float tile[16][132];  // pad 128->132 to dodge bank conflicts
  const int m0 = blockIdx.x * 16;
  const int tid = threadIdx.x;

  {  // cooperative load of the 16x128 A tile
    int r  = tid >> 4;         // 0..15
    int c0 = (tid & 15) * 8;   // 0..120
    int row = m0 + r; if (row >= N) row = N - 1;
    const float* src = X + (long long)row * 128 + c0;
    float4 a = *(const float4*)(src);
    float4 b = *(const float4*)(src + 4);
    tile[r][c0 + 0] = a.x; tile[r][c0 + 1] = a.y;
    tile[r][c0 + 2] = a.z; tile[r][c0 + 3] = a.w;
    tile[r][c0 + 4] = b.x; tile[r][c0 + 5] = b.y;
    tile[r][c0 + 6] = b.z; tile[r][c0 + 7] = b.w;
  }
  __syncthreads();

  const int wave = tid >> 5;       // 0..7 -> output col tile
  const int lane = tid & 31;
  const int half = lane >> 4;      // 0: K=k0,k0+1  1: K=k0+2,k0+3
  const int l    = lane & 15;
  const int n0   = wave * 16;

  v8f c = {};
#pragma unroll
  for (int s = 0; s < 32; ++s) {
    const int k0 = s * 4;
    v2f a, b;
    a.x = tile[l][k0 + 2 * half];
    a.y = tile[l][k0 + 2 * half + 1];
    const float* wp = W + (long long)(k0 + 2 * half) * 128 + n0 + l;
    b.x = wp[0];
    b.y = wp[128];
    c = __builtin_amdgcn_wmma_f32_16x16x4_f32(false, a, false, b, (short)0, c,
                                              false, false);
  }

  const float bc = bias[n0 + l];
#pragma unroll
  for (int v = 0; v < 8; ++v) {
    int row = m0 + v + 8 * half;
    if (row < N) Y[(long long)row * 128 + n0 + l] = c[v] + bc;
  }
}

// ---------------------------------------------------------------------------
// One wave per edge: scores[e,h] = leakyrelu(dot(Q[src], K[dst]) / sqrt(D)).
// Lane i handles float4 of dims 4i..4i+3 (head = i/8). Coalesced 512B gathers.
// Also reduces a per-head global max (ordered-uint atomic max).
// ---------------------------------------------------------------------------
__global__ __launch_bounds__(256) void scores_kernel(
    const float* __restrict__ Q, const float* __restrict__ K,
    const long long* __restrict__ src, const long long* __restrict__ dst,
    float* __restrict__ scores, unsigned* __restrict__ gmax, int E) {
  __shared__ unsigned smax[HEADS];
  const int tid = threadIdx.x;
  if (tid < HEADS) smax[tid] = 0u;
  __syncthreads();

  const int wave = tid >> 5, lane = tid & 31;
  const int e = blockIdx.x * 8 + wave;
  if (e < E) {
    const long long s = src[e], d = dst[e];
    const float4 q = *(const float4*)(Q + s * 128 + lane * 4);
    const float4 k = *(const float4*)(K + d * 128 + lane * 4);
    float p = q.x * k.x + q.y * k.y + q.z * k.z + q.w * k.w;
    p += __shfl_xor(p, 1, 32);
    p += __shfl_xor(p, 2, 32);
    p += __shfl_xor(p, 4, 32);   // 8-lane (per-head) reduction
    if ((lane & 7) == 0) {
      const int h = lane >> 3;
      float sc = p * RSQRT_D;
      sc = sc >= 0.0f ? sc : NEG_SLOPE * sc;  // LeakyReLU(0.2)
      scores[(long long)e * HEADS + h] = sc;
      atomicMax(&smax[h], fmap(sc));
    }
  }
  __syncthreads();
  if (tid < HEADS) atomicMax(&gmax[tid], smax[tid]);
}

// ---------------------------------------------------------------------------
// scores[i] <- exp(scores[i] - max[h]); per-head global sum via LDS + atomics.
// ---------------------------------------------------------------------------
__global__ __launch_bounds__(256) void expsum_kernel(
    float* __restrict__ scores, const unsigned* __restrict__ gmax,
    float* __restrict__ sums, long long total) {
  __shared__ float ssum[HEADS];
  if (threadIdx.x < HEADS) ssum[threadIdx.x] = 0.0f;
  __syncthreads();
  const long long i = (long long)blockIdx.x * blockDim.x + threadIdx.x;
  if (i < total) {
    const int h = (int)(i & (HEADS - 1));
    const float m = funmap(gmax[h]);
    const float ev = __expf(scores[i] - m);
    scores[i] = ev;
    atomicAdd(&ssum[h], ev);
  }
  __syncthreads();
  if (threadIdx.x < HEADS) atomicAdd(&sums[threadIdx.x], ssum[threadIdx.x]);
}

// ---------------------------------------------------------------------------
// One wave per edge: acc[dst] += (exp/sum) * V[src]; global_atomic_add_f32.
// ---------------------------------------------------------------------------
__global__ __launch_bounds__(256) void scatter_kernel(
    const float* __restrict__ V, const float* __restrict__ exps,
    const long long* __restrict__ src, const long long* __restrict__ dst,
    const float* __restrict__ sums, float* __restrict__ acc, int E) {
  const int tid = threadIdx.x;
  const int wave = tid >> 5, lane = tid & 31;
  const int e = blockIdx.x * 8 + wave;
  if (e >= E) return;
  const long long s = src[e], d = dst[e];
  const int h = lane >> 3;
  const float w = exps[(long long)e * HEADS + h] / sums[h];
  const float4 v = *(const float4*)(V + s * 128 + lane * 4);
  float* out = acc + d * 128 + lane * 4;
  atomicAdd(out + 0, v.x * w);
  atomicAdd(out + 1, v.y * w);
  atomicAdd(out + 2, v.z * w);
  atomicAdd(out + 3, v.w * w);
}

// ---------------------------------------------------------------------------
extern "C" void kernel_launch(void* const* d_in, const int* in_sizes, int n_in,
                              void* d_out, int out_size, void* d_ws, size_t ws_size,
                              hipStream_t stream) {
  const float*     x  = (const float*)d_in[0];
  const long long* ei = (const long long*)d_in[1];  // int64 (2,E): [src | dst]
  const float* Wq = (const float*)d_in[2];
  const float* bq = (const float*)d_in[3];
  const float* Wk = (const float*)d_in[4];
  const float* bk = (const float*)d_in[5];
  const float* Wv = (const float*)d_in[6];
  const float* bv = (const float*)d_in[7];
  const float* Wo = (const float*)d_in[8];
  const float* bo = (const float*)d_in[9];

  const int N = in_sizes[0] / 128;
  const int E = in_sizes[1] / 2;
  const long long* src = ei;
  const long long* dst = ei + E;

  char* ws = (char*)d_ws;
  float* Q      = (float*)ws;  ws += (size_t)N * 128 * sizeof(float);
  float* K      = (float*)ws;  ws += (size_t)N * 128 * sizeof(float);
  float* Vv     = (float*)ws;  ws += (size_t)N * 128 * sizeof(float);
  float* scores = (float*)ws;  ws += (size_t)E * HEADS * sizeof(float);
  unsigned* gmax = (unsigned*)ws; ws += HEADS * sizeof(unsigned);
  float* sums   = (float*)ws;
  float* acc    = (float*)d_out;  // accumulator aliased with output

  init_kernel<<<1024, 256, 0, stream>>>(acc, (long long)N * 128, gmax, sums);

  const int mt = (N + 15) / 16;
  gemm128_wmma<<<mt, 256, 0, stream>>>(x, Wq, bq, Q,  N);
  gemm128_wmma<<<mt, 256, 0, stream>>>(x, Wk, bk, K,  N);
  gemm128_wmma<<<mt, 256, 0, stream>>>(x, Wv, bv, Vv, N);

  const int eb = (E + 7) / 8;
  scores_kernel<<<eb, 256, 0, stream>>>(Q, K, src, dst, scores, gmax, E);

  const long long total = (long long)E * HEADS;
  expsum_kernel<<<(int)((total + 255) / 256), 256, 0, stream>>>(scores, gmax, sums, total);

  scatter_kernel<<<eb, 256, 0, stream>>>(Vv, scores, src, dst, sums, acc, E);

  // In-place output projection: safe because A-tiles are staged through LDS.
  gemm128_wmma<<<mt, 256, 0, stream>>>(acc, Wo, bo, acc, N);
}